// GATv2Layer_1760936591672
// MI455X (gfx1250) — compile-verified
//
#include <hip/hip_runtime.h>
#include <math.h>

#define N_NODES_C 50000
#define N_EDGES_C 625000
#define DIM 128
#define SLOPE_C 0.2f
#define TPAD 132   // padded LDS row stride (floats): (m*132+k)%64 varies with m -> no bank conflicts

typedef __attribute__((ext_vector_type(2))) float v2f;
typedef __attribute__((ext_vector_type(8))) float v8f;

// ---- helpers -------------------------------------------------------------

__device__ __forceinline__ void atomicMaxF(float* addr, float val) {
  // monotone int trick: works with -inf init, mixed signs
  if (val >= 0.f) atomicMax((int*)addr, __float_as_int(val));
  else            atomicMin((unsigned int*)addr, __float_as_uint(val));
}

// Stage a 16xDIM fp32 tile into LDS (row stride TPAD), coalesced float4 loads.
// Rows >= nrows are zero-filled. One wave (32 lanes) does the whole tile.
__device__ __forceinline__ void load_tile16x128(float* xt, const float* __restrict__ src,
                                                long long row0, int nrows, int lane) {
#pragma unroll
  for (int j = 0; j < 16; ++j) {
    float4 v = make_float4(0.f, 0.f, 0.f, 0.f);
    if (j < nrows) v = *(const float4*)(src + (size_t)(row0 + j) * DIM + lane * 4);
    *(float4*)(xt + j * TPAD + lane * 4) = v;
  }
}

// One wave computes Out[16][128] = X(16x128, in LDS) @ W^T + bias using
// V_WMMA_F32_16X16X4_F32.  Wp is the k-pair-major repacked weight:
//   Wp[kp*128 + n] = ( W[n][2*kp], W[n][2*kp+1] )   (float2)
// so the B-operand load for lane (col,half) is contiguous across lanes.
__device__ __forceinline__ void wave_gemm_16x128(const float* xt,
                                                 const float2* __restrict__ Wp,
                                                 const float* __restrict__ bias,
                                                 v8f acc[8], int col, int half) {
#pragma unroll
  for (int nt = 0; nt < 8; ++nt) {
    float bv = bias[nt * 16 + col];
    v8f t;
#pragma unroll
    for (int i = 0; i < 8; ++i) t[i] = bv;
    acc[nt] = t;
  }
  // A layout: lane = (k/2)*16 + m, reg = k&1  ->  this lane holds pair kp = k0/2 + half of row col
  // B layout: lane = (k/2)*16 + n, reg = k&1  ->  B[k][n] = W[n][k] = Wp[kp][n]
  for (int k0 = 0; k0 < DIM; k0 += 4) {
    int kp = (k0 >> 1) + half;
    v2f a;
    a.x = xt[col * TPAD + kp * 2];
    a.y = xt[col * TPAD + kp * 2 + 1];
#pragma unroll
    for (int nt = 0; nt < 8; ++nt) {
      float2 bw = Wp[(size_t)kp * DIM + nt * 16 + col];
      v2f b;
      b.x = bw.x;
      b.y = bw.y;
      acc[nt] = __builtin_amdgcn_wmma_f32_16x16x4_f32(
          false, a, false, b, (short)0, acc[nt], false, false);
    }
  }
}

// ---- kernels -------------------------------------------------------------

__global__ void k_init(float* mbuf, float* denom, float* sum_alpha) {
  int i = blockIdx.x * blockDim.x + threadIdx.x;
  if (i < N_NODES_C) {
    mbuf[i] = __int_as_float(0xFF800000);  // -inf
    denom[i] = 0.f;
    sum_alpha[i] = 0.f;
  }
}

// Repack 4 weight matrices (row-major [n][k]) into k-pair-major float2 layout:
//   Wp[kp*128 + n] = { W[n][2kp], W[n][2kp+1] }
__global__ void k_pack_w(const float* __restrict__ Wq, const float* __restrict__ Wk,
                         const float* __restrict__ Wv, const float* __restrict__ We,
                         float2* __restrict__ Wqp, float2* __restrict__ Wkp,
                         float2* __restrict__ Wvp, float2* __restrict__ Wep) {
  int i = blockIdx.x * blockDim.x + threadIdx.x;
  const int per = (DIM / 2) * DIM;  // 8192 float2 per weight
  if (i >= 4 * per) return;
  int w = i / per;
  int idx = i % per;
  int kp = idx >> 7;   // 0..63
  int n  = idx & 127;  // 0..127
  const float* W = (w == 0) ? Wq : (w == 1) ? Wk : (w == 2) ? Wv : We;
  float2* O = (w == 0) ? Wqp : (w == 1) ? Wkp : (w == 2) ? Wvp : Wep;
  float2 v;
  v.x = W[(size_t)n * DIM + kp * 2];
  v.y = W[(size_t)n * DIM + kp * 2 + 1];
  O[idx] = v;
}

// Per-node projections: Qn = V Wq^T + bq, Kn = V Wk^T + bk.
// job = (node-tile, which-weight); one wave per job.
__global__ void k_node_proj(const float* __restrict__ V,
                            const float2* __restrict__ Wqp, const float* __restrict__ bq,
                            const float2* __restrict__ Wkp, const float* __restrict__ bk,
                            float* __restrict__ Qn, float* __restrict__ Kn) {
  __shared__ float lds[4][16 * TPAD];
  int wave = threadIdx.x >> 5, lane = threadIdx.x & 31;
  int job = blockIdx.x * 4 + wave;
  const int njobs = 2 * (N_NODES_C / 16);
  if (job >= njobs) return;
  int tile = job >> 1, which = job & 1;
  int col = lane & 15, half = lane >> 4;

  float* xt = lds[wave];
  load_tile16x128(xt, V, (long long)tile * 16, 16, lane);

  const float2* Wp = which ? Wkp : Wqp;
  const float* b = which ? bk : bq;
  float* out = which ? Kn : Qn;

  v8f acc[8];
  wave_gemm_16x128(xt, Wp, b, acc, col, half);

#pragma unroll
  for (int nt = 0; nt < 8; ++nt)
#pragma unroll
    for (int r = 0; r < 8; ++r) {
      int m = r + 8 * half;
      out[(size_t)(tile * 16 + m) * DIM + nt * 16 + col] = acc[nt][r];
    }
}

// Per-edge: bias = E We^T + be (WMMA), z = leaky(bias + Qn[src] + Kn[dst]),
// score = a.z + ab; store scores; atomic-max into per-dest running max.
__global__ void k_edge_score(const float* __restrict__ E,
                             const float2* __restrict__ Wep, const float* __restrict__ be,
                             const int* __restrict__ ei,
                             const float* __restrict__ Qn, const float* __restrict__ Kn,
                             const float* __restrict__ aw, const float* __restrict__ ab,
                             float* __restrict__ scores, float* __restrict__ mbuf) {
  __shared__ float lds[4][16 * TPAD];
  int wave = threadIdx.x >> 5, lane = threadIdx.x & 31;
  long long tile = (long long)blockIdx.x * 4 + wave;
  const long long ntiles = (N_EDGES_C + 15) / 16;
  if (tile >= ntiles) return;
  long long e0 = tile * 16;
  long long rem = (long long)N_EDGES_C - e0;
  int cnt = rem < 16 ? (int)rem : 16;
  int col = lane & 15, half = lane >> 4;

  float* xt = lds[wave];
  load_tile16x128(xt, E, e0, cnt, lane);

  const int* srcI = ei;
  const int* dstI = ei + N_EDGES_C;
  int srcs[8], dsts[8];
#pragma unroll
  for (int r = 0; r < 8; ++r) {
    long long e = e0 + r + 8 * half;
    bool ok = e < N_EDGES_C;
    srcs[r] = ok ? srcI[e] : 0;
    dsts[r] = ok ? dstI[e] : 0;
  }

  v8f acc[8];
  wave_gemm_16x128(xt, Wep, be, acc, col, half);

  float p[8];
#pragma unroll
  for (int r = 0; r < 8; ++r) p[r] = 0.f;

#pragma unroll
  for (int nt = 0; nt < 8; ++nt) {
    float an = aw[nt * 16 + col];
#pragma unroll
    for (int r = 0; r < 8; ++r) {
      float q = Qn[(size_t)srcs[r] * DIM + nt * 16 + col];
      float kv = Kn[(size_t)dsts[r] * DIM + nt * 16 + col];
      float z = acc[nt][r] + q + kv;
      z = z > 0.f ? z : SLOPE_C * z;
      p[r] = fmaf(an, z, p[r]);
    }
  }

  float abv = ab[0];
#pragma unroll
  for (int r = 0; r < 8; ++r) {
    float s = p[r];
    s += __shfl_xor(s, 1, 16);
    s += __shfl_xor(s, 2, 16);
    s += __shfl_xor(s, 4, 16);
    s += __shfl_xor(s, 8, 16);
    if (col == 0) {
      long long e = e0 + r + 8 * half;
      if (e < N_EDGES_C) {
        float sc = s + abv;
        scores[e] = sc;
        atomicMaxF(&mbuf[dsts[r]], sc);
      }
    }
  }
}

__global__ void k_edge_exp(const int* __restrict__ ei, const float* __restrict__ mbuf,
                           float* __restrict__ scores, float* __restrict__ denom) {
  int e = blockIdx.x * blockDim.x + threadIdx.x;
  if (e >= N_EDGES_C) return;
  int d = ei[N_EDGES_C + e];
  float ev = __expf(scores[e] - mbuf[d]);
  scores[e] = ev;  // overwrite with exp value
  atomicAdd(&denom[d], ev);
}

__global__ void k_edge_alpha(const int* __restrict__ ei, const float* __restrict__ denom,
                             const float* __restrict__ scores, float* __restrict__ sum_alpha) {
  int e = blockIdx.x * blockDim.x + threadIdx.x;
  if (e >= N_EDGES_C) return;
  int d = ei[N_EDGES_C + e];
  float alpha = scores[e] / denom[d];
  atomicAdd(&sum_alpha[d], alpha);
}

// H[n] = sum_alpha[n] * (V[n] Wv^T + bv)   (exact: Vv depends only on dest)
__global__ void k_out(const float* __restrict__ V,
                      const float2* __restrict__ Wvp, const float* __restrict__ bv,
                      const float* __restrict__ sum_alpha, float* __restrict__ H) {
  __shared__ float lds[4][16 * TPAD];
  int wave = threadIdx.x >> 5, lane = threadIdx.x & 31;
  int tile = blockIdx.x * 4 + wave;
  if (tile >= N_NODES_C / 16) return;
  int col = lane & 15, half = lane >> 4;

  float* xt = lds[wave];
  load_tile16x128(xt, V, (long long)tile * 16, 16, lane);

  v8f acc[8];
  wave_gemm_16x128(xt, Wvp, bv, acc, col, half);

  float sr[8];
#pragma unroll
  for (int r = 0; r < 8; ++r) sr[r] = sum_alpha[tile * 16 + r + 8 * half];

#pragma unroll
  for (int nt = 0; nt < 8; ++nt)
#pragma unroll
    for (int r = 0; r < 8; ++r) {
      int m = r + 8 * half;
      H[(size_t)(tile * 16 + m) * DIM + nt * 16 + col] = sr[r] * acc[nt][r];
    }
}

// ---- launch --------------------------------------------------------------

extern "C" void kernel_launch(void* const* d_in, const int* in_sizes, int n_in,
                              void* d_out, int out_size, void* d_ws, size_t ws_size,
                              hipStream_t stream) {
  const float* V  = (const float*)d_in[0];
  const float* E  = (const float*)d_in[1];
  const int*   ei = (const int*)d_in[2];   // [2, N_EDGES]: src row 0, dest row 1
  const float* Wq = (const float*)d_in[3];
  const float* bq = (const float*)d_in[4];
  const float* Wk = (const float*)d_in[5];
  const float* bk = (const float*)d_in[6];
  const float* Wv = (const float*)d_in[7];
  const float* bv = (const float*)d_in[8];
  const float* We = (const float*)d_in[9];
  const float* be = (const float*)d_in[10];
  const float* aw = (const float*)d_in[11];
  const float* ab = (const float*)d_in[12];
  float* H = (float*)d_out;

  float* ws = (float*)d_ws;
  float* Qn        = ws;                                  // [N, 128]
  float* Kn        = Qn + (size_t)N_NODES_C * DIM;        // [N, 128]
  float* scores    = Kn + (size_t)N_NODES_C * DIM;        // [NE]
  float* mbuf      = scores + N_EDGES_C;                  // [N]
  float* denom     = mbuf + N_NODES_C;                    // [N]
  float* sum_alpha = denom + N_NODES_C;                   // [N]
  float2* Wqp      = (float2*)(sum_alpha + N_NODES_C);    // [64*128] float2 each
  float2* Wkp      = Wqp + (DIM / 2) * DIM;
  float2* Wvp      = Wkp + (DIM / 2) * DIM;
  float2* Wep      = Wvp + (DIM / 2) * DIM;

  const int nodeTiles = N_NODES_C / 16;                   // 3125
  const int projJobs  = 2 * nodeTiles;                    // 6250 waves
  const long long edgeTiles = (N_EDGES_C + 15) / 16;      // 39063 waves
  const int packElems = 4 * (DIM / 2) * DIM;              // 32768

  k_init<<<(N_NODES_C + 255) / 256, 256, 0, stream>>>(mbuf, denom, sum_alpha);
  k_pack_w<<<(packElems + 255) / 256, 256, 0, stream>>>(Wq, Wk, Wv, We, Wqp, Wkp, Wvp, Wep);
  k_node_proj<<<(projJobs + 3) / 4, 128, 0, stream>>>(V, Wqp, bq, Wkp, bk, Qn, Kn);
  k_edge_score<<<(int)((edgeTiles + 3) / 4), 128, 0, stream>>>(E, Wep, be, ei, Qn, Kn,
                                                               aw, ab, scores, mbuf);
  k_edge_exp<<<(N_EDGES_C + 255) / 256, 256, 0, stream>>>(ei, mbuf, scores, denom);
  k_edge_alpha<<<(N_EDGES_C + 255) / 256, 256, 0, stream>>>(ei, denom, scores, sum_alpha);
  k_out<<<(nodeTiles + 3) / 4, 128, 0, stream>>>(V, Wvp, bv, sum_alpha, H);
}